// SimpleGCN_21466246546229
// MI455X (gfx1250) — compile-verified
//
#include <hip/hip_runtime.h>

typedef __attribute__((ext_vector_type(2))) float v2f;
typedef __attribute__((ext_vector_type(8))) float v8f;

// ---------------------------------------------------------------- atomics
static __device__ __forceinline__ void atomic_add_f32g(float* p, float v) {
  // agent-scope relaxed fp32 add -> global_atomic_add_f32 (no CAS loop)
  __hip_atomic_fetch_add(p, v, __ATOMIC_RELAXED, __HIP_MEMORY_SCOPE_AGENT);
}

// ---------------------------------------------------------------- utility
__global__ void zero_f32(float* __restrict__ p, int n) {
  int i = blockIdx.x * blockDim.x + threadIdx.x;
  if (i < n) p[i] = 0.0f;
}

__global__ void deg_accum(const int* __restrict__ dst, float* __restrict__ deg, int E) {
  int e = blockIdx.x * blockDim.x + threadIdx.x;
  if (e < E) atomic_add_f32g(&deg[dst[e]], 1.0f);
}

__global__ void deg_finalize(float* __restrict__ dii, int N) {
  int i = blockIdx.x * blockDim.x + threadIdx.x;
  if (i < N) dii[i] = rsqrtf(dii[i] + 1.0f);
}

// ---------------------------------------------------------------- WMMA GEMM
// C[M x 64] = A[M x K] * W[K x 64], row-major, one wave -> one 16x16 tile.
// Uses V_WMMA_F32_16X16X4_F32 (fp32 in, fp32 acc) - exact fp32 GEMM.
template <int K>
__global__ __launch_bounds__(256) void gemm_wmma_f32(
    const float* __restrict__ A, const float* __restrict__ W,
    float* __restrict__ C, int M) {
  const int lane = threadIdx.x & 31;
  const int gw = blockIdx.x * 8 + (threadIdx.x >> 5);   // 8 waves / block
  const int mtile = gw >> 2;                            // 64/16 = 4 n-tiles
  const int ntile = gw & 3;
  if (mtile * 16 >= M) return;                          // wave-uniform guard
  const bool hi = lane >= 16;
  const int l16 = lane & 15;
  const size_t arow = (size_t)(mtile * 16 + l16) * K;
  const int ncol = ntile * 16 + l16;
  const int koff = hi ? 2 : 0;
  v8f acc = {};
#pragma unroll
  for (int k = 0; k < K; k += 4) {
    // A 16x4 frag: lanes 0-15 hold K=k..k+1, lanes 16-31 hold K=k+2..k+3
    v2f a = *(const v2f*)(A + arow + (k + koff));
    v2f b;
    b.x = W[(size_t)(k + koff) * 64 + ncol];
    b.y = W[(size_t)(k + koff + 1) * 64 + ncol];
    acc = __builtin_amdgcn_wmma_f32_16x16x4_f32(false, a, false, b,
                                                (short)0, acc, false, false);
  }
  // C/D layout: VGPR r -> row M=r (lanes 0-15) / M=r+8 (lanes 16-31)
  const int rowbase = mtile * 16 + (hi ? 8 : 0);
#pragma unroll
  for (int r = 0; r < 8; ++r)
    C[(size_t)(rowbase + r) * 64 + ncol] = acc[r];
}

// ---------------------------------------------------------------- GCN pieces
// agg[n][:] = h_lin[n][:] * dii[n]^2 + b[:]   (self edge + bias), H=64
__global__ void gcn_self_init(const float* __restrict__ hlin,
                              const float* __restrict__ dii,
                              const float* __restrict__ bias,
                              float* __restrict__ agg, int N) {
  int gid = blockIdx.x * blockDim.x + threadIdx.x;
  int node = gid >> 4;
  int c = (gid & 15) * 4;
  if (node >= N) return;
  float d2 = dii[node];
  d2 *= d2;
  float4 v = *(const float4*)(hlin + (size_t)node * 64 + c);
  float4 b = *(const float4*)(bias + c);
  float4 r;
  r.x = v.x * d2 + b.x; r.y = v.y * d2 + b.y;
  r.z = v.z * d2 + b.z; r.w = v.w * d2 + b.w;
  *(float4*)(agg + (size_t)node * 64 + c) = r;
}

// agg[dst][:] += h_lin[src][:] * dii[src]*dii[dst];  16 lanes per edge, float4
__global__ __launch_bounds__(256) void edge_scatter64(
    const int* __restrict__ src, const int* __restrict__ dst,
    const float* __restrict__ dii, const float* __restrict__ hlin,
    float* __restrict__ agg, int E) {
  long gid = (long)blockIdx.x * blockDim.x + threadIdx.x;
  long e = gid >> 4;
  int c = ((int)gid & 15) * 4;
  if (e >= E) return;
  int s = src[e], d = dst[e];
  float nrm = dii[s] * dii[d];
  float4 v = *(const float4*)(hlin + (size_t)s * 64 + c);
  float* p = agg + (size_t)d * 64 + c;
  atomic_add_f32g(p + 0, v.x * nrm);
  atomic_add_f32g(p + 1, v.y * nrm);
  atomic_add_f32g(p + 2, v.z * nrm);
  atomic_add_f32g(p + 3, v.w * nrm);
}

// out[n][:] = relu(layernorm(in[n][:]) * g + be); one wave32 per node, H=64
__global__ __launch_bounds__(256) void ln_relu(
    const float* __restrict__ in, float* __restrict__ out,
    const float* __restrict__ g, const float* __restrict__ be, int N) {
  int lane = threadIdx.x & 31;
  int node = blockIdx.x * 8 + (threadIdx.x >> 5);
  if (node >= N) return;
  float2 v = *(const float2*)(in + (size_t)node * 64 + lane * 2);
  float s = v.x + v.y;
#pragma unroll
  for (int o = 16; o; o >>= 1) s += __shfl_xor(s, o, 32);
  float mu = s * (1.0f / 64.0f);
  float dx = v.x - mu, dy = v.y - mu;
  float q = dx * dx + dy * dy;
#pragma unroll
  for (int o = 16; o; o >>= 1) q += __shfl_xor(q, o, 32);
  float inv = rsqrtf(q * (1.0f / 64.0f) + 1e-5f);
  float2 gg = *(const float2*)(g + lane * 2);
  float2 bb = *(const float2*)(be + lane * 2);
  float2 r;
  r.x = fmaxf(dx * inv * gg.x + bb.x, 0.0f);
  r.y = fmaxf(dy * inv * gg.y + bb.y, 0.0f);
  *(float2*)(out + (size_t)node * 64 + lane * 2) = r;
}

// h3lin[n] = dot(h[n][:], w[:]); one wave per node
__global__ __launch_bounds__(256) void gemv_h_to_1(
    const float* __restrict__ h, const float* __restrict__ w,
    float* __restrict__ out, int N) {
  int lane = threadIdx.x & 31;
  int node = blockIdx.x * 8 + (threadIdx.x >> 5);
  if (node >= N) return;
  float2 v = *(const float2*)(h + (size_t)node * 64 + lane * 2);
  float2 ww = *(const float2*)(w + lane * 2);
  float s = v.x * ww.x + v.y * ww.y;
#pragma unroll
  for (int o = 16; o; o >>= 1) s += __shfl_xor(s, o, 32);
  if (lane == 0) out[node] = s;
}

__global__ void gcn_self_init1(const float* __restrict__ h3,
                               const float* __restrict__ dii,
                               const float* __restrict__ b3,
                               float* __restrict__ out, int N) {
  int i = blockIdx.x * blockDim.x + threadIdx.x;
  if (i < N) {
    float d = dii[i];
    out[i] = h3[i] * d * d + b3[0];
  }
}

__global__ void edge_scatter1(const int* __restrict__ src,
                              const int* __restrict__ dst,
                              const float* __restrict__ dii,
                              const float* __restrict__ h3,
                              float* __restrict__ out, int E) {
  int e = blockIdx.x * blockDim.x + threadIdx.x;
  if (e < E) {
    int s = src[e], d = dst[e];
    atomic_add_f32g(&out[d], h3[s] * dii[s] * dii[d]);
  }
}

// mean-pool via LDS buckets (ds_add_f32), one flush of G atomics per block
__global__ __launch_bounds__(256) void pool_kernel(
    const float* __restrict__ vals, const int* __restrict__ batch,
    float* __restrict__ sums, float* __restrict__ cnts, int N, int G) {
  __shared__ float ssum[64];
  __shared__ float scnt[64];
  int t = threadIdx.x;
  if (t < 64) { ssum[t] = 0.0f; scnt[t] = 0.0f; }
  __syncthreads();
  int i = blockIdx.x * blockDim.x + t;
  if (i < N) {
    int g = batch[i] & 63;
    atomicAdd(&ssum[g], vals[i]);   // ds_add_f32
    atomicAdd(&scnt[g], 1.0f);
  }
  __syncthreads();
  if (t < G && t < 64) {
    if (scnt[t] != 0.0f) {
      atomic_add_f32g(&sums[t], ssum[t]);
      atomic_add_f32g(&cnts[t], scnt[t]);
    }
  }
}

__global__ void graph_final(const float* __restrict__ sums,
                            const float* __restrict__ cnts,
                            const float* __restrict__ Wg,
                            const float* __restrict__ bg,
                            float* __restrict__ gout, int G) {
  int g = blockIdx.x * blockDim.x + threadIdx.x;
  if (g < G) gout[g] = (sums[g] / fmaxf(cnts[g], 1.0f)) * Wg[0] + bg[0];
}

// ---------------------------------------------------------------- launcher
extern "C" void kernel_launch(void* const* d_in, const int* in_sizes, int n_in,
                              void* d_out, int out_size, void* d_ws, size_t ws_size,
                              hipStream_t stream) {
  const float* x   = (const float*)d_in[0];
  const int*   ei  = (const int*)d_in[1];
  const int* batch = (const int*)d_in[2];
  const float* W1  = (const float*)d_in[3];
  const float* b1  = (const float*)d_in[4];
  const float* g1  = (const float*)d_in[5];
  const float* be1 = (const float*)d_in[6];
  const float* W2  = (const float*)d_in[7];
  const float* b2  = (const float*)d_in[8];
  const float* g2  = (const float*)d_in[9];
  const float* be2 = (const float*)d_in[10];
  const float* W3  = (const float*)d_in[11];
  const float* b3  = (const float*)d_in[12];
  const float* Wg  = (const float*)d_in[13];
  const float* bg  = (const float*)d_in[14];

  const int H   = in_sizes[4];            // 64
  const int IN_ = in_sizes[3] / H;        // 128
  const int N   = in_sizes[0] / IN_;      // 100000
  const int E   = in_sizes[1] / 2;        // 1600000
  const int G   = out_size - N;           // 64
  const int* src = ei;
  const int* dst = ei + E;

  // workspace carve-up (floats)
  float* w     = (float*)d_ws;
  float* dii   = w;                         // N
  float* h3lin = w + N;                     // N
  float* sums  = w + 2 * (size_t)N;         // G
  float* cnts  = sums + G;                  // G
  size_t off   = ((2 * (size_t)N + 2 * (size_t)G) + 63) & ~(size_t)63;
  float* bufA  = w + off;                   // N*64
  float* bufB  = bufA + (size_t)N * 64;     // N*64

  float* out_nodes = (float*)d_out;
  float* out_graph = out_nodes + N;

  const int B = 256;
  const int mtiles = (N + 15) / 16;
  const int gemm_blocks = (mtiles * 4 + 7) / 8;
  const int node_waves_blocks = (N + 7) / 8;
  const unsigned sc64_blocks = (unsigned)(((size_t)E * 16 + B - 1) / B);
  const unsigned self_blocks = (unsigned)(((size_t)N * 16 + B - 1) / B);

  // degrees
  zero_f32<<<(N + B - 1) / B, B, 0, stream>>>(dii, N);
  deg_accum<<<(E + B - 1) / B, B, 0, stream>>>(dst, dii, E);
  deg_finalize<<<(N + B - 1) / B, B, 0, stream>>>(dii, N);

  // layer 1: h1_lin = x @ W1 -> bufA; agg -> bufB; h1 = relu(ln(agg)) -> bufA
  gemm_wmma_f32<128><<<gemm_blocks, B, 0, stream>>>(x, W1, bufA, N);
  gcn_self_init<<<self_blocks, B, 0, stream>>>(bufA, dii, b1, bufB, N);
  edge_scatter64<<<sc64_blocks, B, 0, stream>>>(src, dst, dii, bufA, bufB, E);
  ln_relu<<<node_waves_blocks, B, 0, stream>>>(bufB, bufA, g1, be1, N);

  // layer 2: h2_lin = h1 @ W2 -> bufB; agg -> bufA; h2 -> bufB
  gemm_wmma_f32<64><<<gemm_blocks, B, 0, stream>>>(bufA, W2, bufB, N);
  gcn_self_init<<<self_blocks, B, 0, stream>>>(bufB, dii, b2, bufA, N);
  edge_scatter64<<<sc64_blocks, B, 0, stream>>>(src, dst, dii, bufB, bufA, E);
  ln_relu<<<node_waves_blocks, B, 0, stream>>>(bufA, bufB, g2, be2, N);

  // layer 3 (scalar): h3lin = h2 @ W3; out = gcn(h3lin)
  gemv_h_to_1<<<node_waves_blocks, B, 0, stream>>>(bufB, W3, h3lin, N);
  gcn_self_init1<<<(N + B - 1) / B, B, 0, stream>>>(h3lin, dii, b3, out_nodes, N);
  edge_scatter1<<<(E + B - 1) / B, B, 0, stream>>>(src, dst, dii, h3lin, out_nodes, E);

  // pooling + graph head
  zero_f32<<<1, B, 0, stream>>>(sums, 2 * G);   // sums & cnts contiguous
  pool_kernel<<<(N + B - 1) / B, B, 0, stream>>>(out_nodes, batch, sums, cnts, N, G);
  graph_final<<<1, 64, 0, stream>>>(sums, cnts, Wg, bg, out_graph, G);
}